// FeatureProjector_15693810499596
// MI455X (gfx1250) — compile-verified
//
#include <hip/hip_runtime.h>

typedef __attribute__((ext_vector_type(2))) float v2f;
typedef __attribute__((ext_vector_type(8))) float v8f;

#define B_  2
#define V_  8
#define C_  32
#define H_  256
#define W_  320
#define HW_ (H_ * W_)

__global__ __launch_bounds__(256) void FeatureProjector_kernel(
    const float* __restrict__ feat,    // (B*V, C, H, W)
    const float* __restrict__ proj,    // (B, V, 4, 4)
    const float* __restrict__ depth,   // (B, H*W)
    float* __restrict__ out)           // (B, V, C, H, W)
{
    const int n    = blockIdx.y;                       // b*V + v
    const int b    = n / V_;
    const int p    = blockIdx.x * blockDim.x + threadIdx.x;  // pixel index in [0, HW)
    const int lane = threadIdx.x & 31;

    // ---- per-lane homogeneous vector h = (x*d, y*d, d, 1) ----
    const float xf = (float)(p % W_);
    const float yf = (float)(p / W_);
    const float d  = depth[b * HW_ + p];
    const float h0 = xf * d;
    const float h1 = yf * d;
    const float h2 = d;

    // ---- A operand: P (4x4 projection, rows 0..2 used) padded to 16x4 ----
    // 32-bit A 16x4 layout: lanes 0-15 hold K={0,1}, lanes 16-31 hold K={2,3}; row M = lane&15.
    const int  r  = lane & 15;
    const int  kb = (lane < 16) ? 0 : 2;
    const int  rc = (r < 4) ? r : 0;                  // safe address for padded rows
    const float* pm = proj + n * 16 + rc * 4 + kb;
    const float a0 = pm[0];
    const float a1 = pm[1];
    v2f A;
    A.x = (r < 4) ? a0 : 0.0f;
    A.y = (r < 4) ? a1 : 0.0f;

    // ---- B operands (4x16, K x N): mirror of A layout.
    // WMMA #1 covers pixels 0..15 of the wave, WMMA #2 covers pixels 16..31.
    const int  lo = lane & 15;
    const bool hi = lane >= 16;
    const float t0 = __shfl(h0, lo, 32);
    const float t1 = __shfl(h1, lo, 32);
    const float t2 = __shfl(h2, lo, 32);
    const float u0 = __shfl(h0, lo | 16, 32);
    const float u1 = __shfl(h1, lo | 16, 32);
    const float u2 = __shfl(h2, lo | 16, 32);
    v2f B1; B1.x = hi ? t2 : t0; B1.y = hi ? 1.0f : t1;
    v2f B2; B2.x = hi ? u2 : u0; B2.y = hi ? 1.0f : u1;

    v8f cz = {};
    // D = A(16x4) x B(4x16): rows 0..2 of D are cg_x, cg_y, cg_z for 16 pixels.
    v8f D1 = __builtin_amdgcn_wmma_f32_16x16x4_f32(
        false, A, false, B1, (short)0, cz, false, false);
    v8f D2 = __builtin_amdgcn_wmma_f32_16x16x4_f32(
        false, A, false, B2, (short)0, cz, false, false);

    // ---- redistribute cg back to one-pixel-per-lane ----
    const float e0 = __shfl(D2[0], lo, 32);
    const float e1 = __shfl(D2[1], lo, 32);
    const float e2 = __shfl(D2[2], lo, 32);
    const float cgx = hi ? e0 : D1[0];
    const float cgy = hi ? e1 : D1[1];
    const float cgz = hi ? e2 : D1[2];

    // ---- grid-sample coordinates (algebraically identical to reference) ----
    // ix = (gx+1)*W/2 - 0.5 with gx = X/(W-1)*2 - 1  ==>  ix = X * W/(W-1) - 0.5
    // v_rcp_f32 (1 ulp) instead of the ~10-op IEEE divide: error << 1e-4 pixel.
    const float inv = __builtin_amdgcn_rcpf(cgz);
    const float ix = cgx * inv * ((float)W_ / (float)(W_ - 1)) - 0.5f;
    const float iy = cgy * inv * ((float)H_ / (float)(H_ - 1)) - 0.5f;

    const float fx0 = floorf(ix);
    const float fy0 = floorf(iy);
    const float wx1 = ix - fx0, wy1 = iy - fy0;
    const float wx0 = 1.0f - wx1, wy0 = 1.0f - wy1;

    const float W1 = (float)(W_ - 1), H1 = (float)(H_ - 1);
    const float x0f = fx0, x1f = fx0 + 1.0f;
    const float y0f = fy0, y1f = fy0 + 1.0f;
    const bool vX0 = (x0f >= 0.0f) && (x0f <= W1);
    const bool vX1 = (x1f >= 0.0f) && (x1f <= W1);
    const bool vY0 = (y0f >= 0.0f) && (y0f <= H1);
    const bool vY1 = (y1f >= 0.0f) && (y1f <= H1);

    const int xs0 = (int)fminf(fmaxf(x0f, 0.0f), W1);
    const int xs1 = (int)fminf(fmaxf(x1f, 0.0f), W1);
    const int ys0 = (int)fminf(fmaxf(y0f, 0.0f), H1);
    const int ys1 = (int)fminf(fmaxf(y1f, 0.0f), H1);

    const float w00 = (vX0 && vY0) ? wx0 * wy0 : 0.0f;
    const float w10 = (vX1 && vY0) ? wx1 * wy0 : 0.0f;
    const float w01 = (vX0 && vY1) ? wx0 * wy1 : 0.0f;
    const float w11 = (vX1 && vY1) ? wx1 * wy1 : 0.0f;

    const int o00 = ys0 * W_ + xs0;
    const int o10 = ys0 * W_ + xs1;
    const int o01 = ys1 * W_ + xs0;
    const int o11 = ys1 * W_ + xs1;

    const size_t base = (size_t)n * C_ * HW_;
    const float* __restrict__ fb = feat + base;
    float* __restrict__ ob = out + base + (size_t)p;

    // Feature reads stay resident in the 192MB L2 (168MB tensor); output is
    // streamed with non-temporal stores so it doesn't evict them.
    // unroll 8 -> 32 loads in flight per thread for L2-latency hiding.
#pragma unroll 8
    for (int c = 0; c < C_; ++c) {
        const float* __restrict__ fc = fb + (size_t)c * HW_;
        float val = w00 * fc[o00];
        val = fmaf(w10, fc[o10], val);
        val = fmaf(w01, fc[o01], val);
        val = fmaf(w11, fc[o11], val);
        __builtin_nontemporal_store(val, ob + (size_t)c * HW_);
    }
}

extern "C" void kernel_launch(void* const* d_in, const int* in_sizes, int n_in,
                              void* d_out, int out_size, void* d_ws, size_t ws_size,
                              hipStream_t stream) {
    (void)in_sizes; (void)n_in; (void)out_size; (void)d_ws; (void)ws_size;
    const float* feat  = (const float*)d_in[0];   // (B,V,C,H,W) f32
    const float* proj  = (const float*)d_in[1];   // (B,V,4,4)   f32
    const float* depth = (const float*)d_in[2];   // (B,H*W)     f32
    float* out = (float*)d_out;                   // (B,V,C,H,W) f32

    dim3 grid(HW_ / 256, B_ * V_);
    FeatureProjector_kernel<<<grid, 256, 0, stream>>>(feat, proj, depth, out);
}